// GCN_58007828300297
// MI455X (gfx1250) — compile-verified
//
#include <hip/hip_runtime.h>
#include <math.h>

typedef float v2f __attribute__((ext_vector_type(2)));
typedef float v8f __attribute__((ext_vector_type(8)));

#define NNODES 100000
#define NEDGES 1600000
#define FIN   64
#define FHID  64
#define FOUT  16

// ---------------- degree / normalization ----------------
__global__ void k_deg_init(float* __restrict__ dinv, int n) {
    int i = blockIdx.x * blockDim.x + threadIdx.x;
    if (i < n) dinv[i] = 1.0f;   // self-loop contributes 1 to every degree
}

__global__ void k_deg_count(const int* __restrict__ dst, float* __restrict__ dinv, int e) {
    int i = blockIdx.x * blockDim.x + threadIdx.x;
    if (i < e) atomicAdd(&dinv[dst[i]], 1.0f);
}

__global__ void k_deg_rsqrt(float* __restrict__ dinv, int n) {
    int i = blockIdx.x * blockDim.x + threadIdx.x;
    if (i < n) dinv[i] = rsqrtf(dinv[i]);   // deg >= 1 always (self-loop)
}

// ---------------- fp32 WMMA GEMM: C[M x N] = A[M x K] @ B[K x N] ----------------
// One wave computes one 16x16 tile with K/4 v_wmma_f32_16x16x4_f32 ops.
// A 16x4 layout : lane l (half = l>>4, lm = l&15): VGPR v = A[lm][2*half + v]
// B 4x16 layout : lane l: VGPR v = B[2*half + v][lm]
// C 16x16 layout: lane l: VGPR r = C[r + 8*half][lm]
template<int KDIM, bool RELU>
__global__ void k_gemm_wmma(const float* __restrict__ A, const float* __restrict__ B,
                            float* __restrict__ C, int mtiles, int ntiles, int ncols) {
    int wave = (blockIdx.x * blockDim.x + threadIdx.x) >> 5;
    int lane = threadIdx.x & 31;
    int tm = wave / ntiles;
    int tn = wave % ntiles;
    if (tm >= mtiles) return;           // whole-wave uniform exit: EXEC stays all-1s for WMMA
    int half = lane >> 4;
    int lm   = lane & 15;

    const float* arow = A + (size_t)(tm * 16 + lm) * KDIM;
    v8f acc = {};
    #pragma unroll 4
    for (int k = 0; k < KDIM; k += 4) {
        v2f a, b;
        a.x = arow[k + 2 * half];
        a.y = arow[k + 2 * half + 1];
        if (RELU) { a.x = fmaxf(a.x, 0.0f); a.y = fmaxf(a.y, 0.0f); }
        b.x = B[(k + 2 * half)     * ncols + tn * 16 + lm];
        b.y = B[(k + 2 * half + 1) * ncols + tn * 16 + lm];
        // 8 args: (neg_a, A, neg_b, B, c_mod, C, reuse_a, reuse_b)
        acc = __builtin_amdgcn_wmma_f32_16x16x4_f32(false, a, false, b,
                                                    (short)0, acc, false, false);
    }
    float* crow = C + (size_t)(tm * 16) * ncols + tn * 16 + lm;
    #pragma unroll
    for (int r = 0; r < 8; ++r)
        crow[(size_t)(r + 8 * half) * ncols] = acc[r];
}

// ---------------- aggregation: self-loop + bias init ----------------
template<int F>
__global__ void k_agg_self(const float* __restrict__ xw, const float* __restrict__ dinv,
                           const float* __restrict__ bias, float* __restrict__ out, int n) {
    int t = blockIdx.x * blockDim.x + threadIdx.x;
    if (t >= n * F) return;
    int i = t / F, f = t % F;
    float di = dinv[i];
    out[t] = xw[t] * di * di + bias[f];
}

// ---------------- aggregation: per-edge scatter-add ----------------
// F=64: 32 threads (one wave) per edge, float2 per lane. F=16: 8 threads per edge.
template<int F>
__global__ void k_agg_edges(const int* __restrict__ src, const int* __restrict__ dst,
                            const float* __restrict__ dinv, const float* __restrict__ xw,
                            float* __restrict__ out, int e) {
    const int PER = F / 2;
    int t = blockIdx.x * blockDim.x + threadIdx.x;
    if (t >= e * PER) return;
    int ei = t / PER;
    int f  = (t % PER) * 2;
    int s = src[ei], d = dst[ei];
    float w = dinv[s] * dinv[d];
    float2 v = *reinterpret_cast<const float2*>(xw + (size_t)s * F + f);
    float* o = out + (size_t)d * F + f;
    atomicAdd(o,     v.x * w);
    atomicAdd(o + 1, v.y * w);
}

// ---------------- row-wise log_softmax over 16 classes ----------------
__global__ void k_logsoftmax16(const float* __restrict__ in, float* __restrict__ out, int n) {
    int i = blockIdx.x * blockDim.x + threadIdx.x;
    if (i >= n) return;
    const float4* p = reinterpret_cast<const float4*>(in + (size_t)i * 16);
    float4 q[4] = { p[0], p[1], p[2], p[3] };
    float* v = reinterpret_cast<float*>(q);
    float m = v[0];
    #pragma unroll
    for (int k = 1; k < 16; ++k) m = fmaxf(m, v[k]);
    float s = 0.0f;
    #pragma unroll
    for (int k = 0; k < 16; ++k) s += __expf(v[k] - m);
    float ls = __logf(s);
    #pragma unroll
    for (int k = 0; k < 16; ++k) v[k] = v[k] - m - ls;
    float4* o = reinterpret_cast<float4*>(out + (size_t)i * 16);
    o[0] = q[0]; o[1] = q[1]; o[2] = q[2]; o[3] = q[3];
}

extern "C" void kernel_launch(void* const* d_in, const int* in_sizes, int n_in,
                              void* d_out, int out_size, void* d_ws, size_t ws_size,
                              hipStream_t stream) {
    const float* x   = (const float*)d_in[0];                 // [N, 64]
    const int*   ei  = (const int*)  d_in[1];                 // [2, E]
    const float* W1  = (const float*)d_in[2];                 // [64, 64]
    const float* b1  = (const float*)d_in[3];                 // [64]
    const float* W2  = (const float*)d_in[4];                 // [64, 16]
    const float* b2  = (const float*)d_in[5];                 // [16]
    float*       out = (float*)d_out;                         // [N, 16]

    const int* src = ei;
    const int* dst = ei + NEDGES;

    // workspace layout (floats), offsets kept 128B-aligned
    float* ws   = (float*)d_ws;
    float* dinv = ws;                                  // N
    float* xw1  = ws  + 100096;                        // N*64
    float* h1   = xw1 + (size_t)NNODES * FHID;         // N*64
    float* xw2  = h1  + (size_t)NNODES * FHID;         // N*16
    float* o2   = xw2 + (size_t)NNODES * FOUT;         // N*16
    // total ~64.4 MB

    const int B = 256;

    // 1) degrees -> dinv
    k_deg_init <<<(NNODES + B - 1) / B, B, 0, stream>>>(dinv, NNODES);
    k_deg_count<<<(NEDGES + B - 1) / B, B, 0, stream>>>(dst, dinv, NEDGES);
    k_deg_rsqrt<<<(NNODES + B - 1) / B, B, 0, stream>>>(dinv, NNODES);

    // 2) xw1 = x @ W1   (6250 x 4 tiles of 16x16, 8 waves/block)
    {
        int mtiles = NNODES / 16, ntiles = FHID / 16;
        int waves = mtiles * ntiles;
        k_gemm_wmma<FIN, false><<<(waves + 7) / 8, B, 0, stream>>>(x, W1, xw1, mtiles, ntiles, FHID);
    }

    // 3) h1 = dinv^2 * xw1 + b1, then scatter-add edge messages
    k_agg_self<FHID> <<<((NNODES * FHID) + B - 1) / B, B, 0, stream>>>(xw1, dinv, b1, h1, NNODES);
    k_agg_edges<FHID><<<((NEDGES * (FHID / 2)) + B - 1) / B, B, 0, stream>>>(src, dst, dinv, xw1, h1, NEDGES);

    // 4) xw2 = relu(h1) @ W2  (ReLU fused into A-matrix load)
    {
        int mtiles = NNODES / 16, ntiles = FOUT / 16;
        int waves = mtiles * ntiles;
        k_gemm_wmma<FHID, true><<<(waves + 7) / 8, B, 0, stream>>>(h1, W2, xw2, mtiles, ntiles, FOUT);
    }

    // 5) o2 = dinv^2 * xw2 + b2, then scatter-add edge messages
    k_agg_self<FOUT> <<<((NNODES * FOUT) + B - 1) / B, B, 0, stream>>>(xw2, dinv, b2, o2, NNODES);
    k_agg_edges<FOUT><<<((NEDGES * (FOUT / 2)) + B - 1) / B, B, 0, stream>>>(src, dst, dinv, xw2, o2, NEDGES);

    // 6) log_softmax per row
    k_logsoftmax16<<<(NNODES + B - 1) / B, B, 0, stream>>>(o2, out, NNODES);
}